// AI4Urban_12902081757452
// MI455X (gfx1250) — compile-verified
//
#include <hip/hip_runtime.h>
#include <hip/hip_bf16.h>
#include <stdint.h>

#define N   128
#define NT  (N * N * N)

#define TX 32
#define TY 8
#define TZ 4
#define LY (TY + 2)               /* 10 */
#define LZ (TZ + 2)               /* 6  */
#define LXP 40                    /* padded row pitch: 160B = 16B-aligned rows   */
#define XOFF 4                    /* gx = x0 lands at lx = 4 (byte 16, aligned)  */
#define TILE_VOL (LXP * LY * LZ)  /* 2400 floats = 9600 B per field              */
#define SXY (LXP * LY)
#define ROWS (LY * LZ)            /* 60 (y,z) rows per tile                      */

__device__ __forceinline__ int clampi(int v, int lo, int hi) {
    return v < lo ? lo : (v > hi ? hi : v);
}

__device__ __forceinline__ float frcp(float x) {
#if __has_builtin(__builtin_amdgcn_rcpf)
    return __builtin_amdgcn_rcpf(x);   // v_rcp_f32
#else
    return 1.0f / x;
#endif
}

__device__ __forceinline__ unsigned lds_dyn_base() {
#if __has_builtin(__builtin_amdgcn_groupstaticsize)
    return (unsigned)__builtin_amdgcn_groupstaticsize();
#else
    return 0u;
#endif
}

// GLOBAL_LOAD_ASYNC_TO_LDS_{B32,B128}, GV mode (64-bit vaddr, saddr=off).
__device__ __forceinline__ void async_ld32(unsigned lds_off, const float* g) {
    asm volatile("global_load_async_to_lds_b32 %0, %1, off"
                 :: "v"(lds_off), "v"((unsigned long long)(uintptr_t)g) : "memory");
}
__device__ __forceinline__ void async_ld128(unsigned lds_off, const float* g) {
    asm volatile("global_load_async_to_lds_b128 %0, %1, off"
                 :: "v"(lds_off), "v"((unsigned long long)(uintptr_t)g) : "memory");
}

__device__ __forceinline__ void wait_async() {
#if __has_builtin(__builtin_amdgcn_s_wait_asynccnt)
    __builtin_amdgcn_s_wait_asynccnt(0);
#else
    asm volatile("s_wait_asynccnt 0x0" ::: "memory");
#endif
}

// Halo tile load, edge-clamped (== edge_pad). Interior rows are 8 x async b128
// (16B aligned both sides, never x-clamped); the two x-halo cells per row are b32.
__device__ __forceinline__ void tile_load_async(const float* __restrict__ g,
                                                int slot, int x0, int y0, int z0,
                                                int tid) {
    const unsigned base = lds_dyn_base() + (unsigned)slot * (unsigned)(TILE_VOL * 4);
    for (int a = tid; a < ROWS * 8; a += 256) {          // divergence-free b128 loop
        int k   = a & 7;
        int row = a >> 3;
        int ly = row % LY, lz = row / LY;
        int gy = clampi(y0 + ly - 1, 0, N - 1);
        int gz = clampi(z0 + lz - 1, 0, N - 1);
        const float* grow = g + (size_t)(gz * N + gy) * N;
        async_ld128(base + (unsigned)(row * LXP + XOFF + 4 * k) * 4u, grow + (x0 + 4 * k));
    }
    for (int h = tid; h < ROWS * 2; h += 256) {          // x halos
        int side = h & 1;
        int row  = h >> 1;
        int ly = row % LY, lz = row / LY;
        int gy = clampi(y0 + ly - 1, 0, N - 1);
        int gz = clampi(z0 + lz - 1, 0, N - 1);
        const float* grow = g + (size_t)(gz * N + gy) * N;
        int gx = side ? clampi(x0 + TX, 0, N - 1) : clampi(x0 - 1, 0, N - 1);
        int lx = side ? (XOFF + TX) : (XOFF - 1);
        async_ld32(base + (unsigned)(row * LXP + lx) * 4u, grow + gx);
    }
}

// Dirichlet inlet (bc_u): padded -x face = ub. Caller must place this between
// barriers (arbitrary thread<->cell ownership).
__device__ __forceinline__ void tile_patch_bcu(float* sm, int slot, int x0,
                                               float ubf, int tid) {
    if (x0 != 0) return;
    float* S = sm + slot * TILE_VOL;
    for (int r = tid; r < ROWS; r += 256) S[r * LXP + (XOFF - 1)] = ubf;
}

__device__ __forceinline__ float lap3(const float* S, int ci) {
    return S[ci - 1] + S[ci + 1] + S[ci - LXP] + S[ci + LXP] +
           S[ci - SXY] + S[ci + SXY] - 6.0f * S[ci];
}
__device__ __forceinline__ float ddx(const float* S, int ci) { return 0.5f * (S[ci + 1]   - S[ci - 1]); }
__device__ __forceinline__ float ddy(const float* S, int ci) { return 0.5f * (S[ci + LXP] - S[ci - LXP]); }
__device__ __forceinline__ float ddz(const float* S, int ci) { return 0.5f * (S[ci + SXY] - S[ci - SXY]); }

// ---- predictor (fused solid filter):
//   uf = u/(1+dt*sig) (also emitted for the corrector),
//   b  = solid(uf + 0.5dt(Re*lap - adv) - dt*grad p)
__global__ void __launch_bounds__(256)
k_predictor(const float* __restrict__ u, const float* __restrict__ v,
            const float* __restrict__ w, const float* __restrict__ p,
            const float* __restrict__ sig, const float* __restrict__ dtp,
            const int* __restrict__ ubp, const int* __restrict__ rep,
            float* __restrict__ bu, float* __restrict__ bv, float* __restrict__ bw,
            float* __restrict__ u0, float* __restrict__ v0, float* __restrict__ w0) {
    extern __shared__ float sm[];
    const int x0 = blockIdx.x * TX, y0 = blockIdx.y * TY, z0 = blockIdx.z * TZ;
    const int tid = threadIdx.y * TX + threadIdx.x;
    tile_load_async(u,   0, x0, y0, z0, tid);
    tile_load_async(v,   1, x0, y0, z0, tid);
    tile_load_async(w,   2, x0, y0, z0, tid);
    tile_load_async(p,   3, x0, y0, z0, tid);
    tile_load_async(sig, 4, x0, y0, z0, tid);
    wait_async();
    __syncthreads();
    const float dt = dtp[0], Re = (float)rep[0];
    const float ubf = (float)ubp[0];
    {   // in-LDS solid filter + bc_u patch; each element written by exactly one thread
        float* Su = sm;
        float* Sv = sm + TILE_VOL;
        float* Sw = sm + 2 * TILE_VOL;
        const float* Ss = sm + 4 * TILE_VOL;
        const bool bcb = (x0 == 0);
        for (int i = tid; i < TILE_VOL; i += 256) {
            float f = frcp(1.0f + dt * Ss[i]);
            bool bc = bcb && ((i % LXP) == (XOFF - 1));
            Su[i] = bc ? ubf : Su[i] * f;
            Sv[i] *= f;
            Sw[i] *= f;
        }
    }
    __syncthreads();
    const float* Su = sm;
    const float* Sv = sm + TILE_VOL;
    const float* Sw = sm + 2 * TILE_VOL;
    const float* Sp = sm + 3 * TILE_VOL;
    const float* Ss = sm + 4 * TILE_VOL;
    const int lx = threadIdx.x + XOFF, ly = threadIdx.y + 1;
    for (int zz = 0; zz < TZ; ++zz) {
        const int ci = ((zz + 1) * LY + ly) * LXP + lx;
        const int gi = ((z0 + zz) * N + (y0 + threadIdx.y)) * N + (x0 + threadIdx.x);
        float uc = Su[ci], vc = Sv[ci], wc = Sw[ci];
        float f = frcp(1.0f + dt * Ss[ci]);
        float bu_ = uc + 0.5f * dt * (Re * lap3(Su, ci) - uc * ddx(Su, ci) - vc * ddy(Su, ci) - wc * ddz(Su, ci)) - dt * ddx(Sp, ci);
        float bv_ = vc + 0.5f * dt * (Re * lap3(Sv, ci) - uc * ddx(Sv, ci) - vc * ddy(Sv, ci) - wc * ddz(Sv, ci)) - dt * ddy(Sp, ci);
        float bw_ = wc + 0.5f * dt * (Re * lap3(Sw, ci) - uc * ddx(Sw, ci) - vc * ddy(Sw, ci) - wc * ddz(Sw, ci)) - dt * ddz(Sp, ci);
        u0[gi] = uc; v0[gi] = vc; w0[gi] = wc;
        bu[gi] = bu_ * f;
        bv[gi] = bv_ * f;
        bw[gi] = bw_ * f;
    }
}

// ---------------- corrector (in-place u,v,w update) ----------------
__global__ void __launch_bounds__(256)
k_corrector(const float* __restrict__ bu, const float* __restrict__ bv,
            const float* __restrict__ bw, const float* __restrict__ p,
            const float* __restrict__ sig, const float* __restrict__ dtp,
            const int* __restrict__ ubp, const int* __restrict__ rep,
            float* __restrict__ u, float* __restrict__ v, float* __restrict__ w) {
    extern __shared__ float sm[];
    const int x0 = blockIdx.x * TX, y0 = blockIdx.y * TY, z0 = blockIdx.z * TZ;
    const int tid = threadIdx.y * TX + threadIdx.x;
    tile_load_async(bu, 0, x0, y0, z0, tid);
    tile_load_async(bv, 1, x0, y0, z0, tid);
    tile_load_async(bw, 2, x0, y0, z0, tid);
    tile_load_async(p,  3, x0, y0, z0, tid);
    wait_async();
    __syncthreads();
    tile_patch_bcu(sm, 0, x0, (float)ubp[0], tid);
    __syncthreads();
    const float dt = dtp[0], Re = (float)rep[0];
    const float* Sa = sm;
    const float* Sb = sm + TILE_VOL;
    const float* Sc = sm + 2 * TILE_VOL;
    const float* Sp = sm + 3 * TILE_VOL;
    const int lx = threadIdx.x + XOFF, ly = threadIdx.y + 1;
    for (int zz = 0; zz < TZ; ++zz) {
        const int ci = ((zz + 1) * LY + ly) * LXP + lx;
        const int gi = ((z0 + zz) * N + (y0 + threadIdx.y)) * N + (x0 + threadIdx.x);
        float buc = Sa[ci], bvc = Sb[ci], bwc = Sc[ci];
        float f = frcp(1.0f + dt * sig[gi]);
        float un = u[gi] + dt * Re * lap3(Sa, ci) - dt * (buc * ddx(Sa, ci) + bvc * ddy(Sa, ci) + bwc * ddz(Sa, ci)) - dt * ddx(Sp, ci);
        float vn = v[gi] + dt * Re * lap3(Sb, ci) - dt * (buc * ddx(Sb, ci) + bvc * ddy(Sb, ci) + bwc * ddz(Sb, ci)) - dt * ddy(Sp, ci);
        float wn = w[gi] + dt * Re * lap3(Sc, ci) - dt * (buc * ddx(Sc, ci) + bvc * ddy(Sc, ci) + bwc * ddz(Sc, ci)) - dt * ddz(Sp, ci);
        u[gi] = un * f;
        v[gi] = vn * f;
        w[gi] = wn * f;
    }
}

// ---------------- divergence rhs: b = -(dx u + dy v + dz w)/dt ----------------
__global__ void __launch_bounds__(256)
k_divergence(const float* __restrict__ u, const float* __restrict__ v,
             const float* __restrict__ w, const float* __restrict__ dtp,
             const int* __restrict__ ubp, float* __restrict__ b) {
    extern __shared__ float sm[];
    const int x0 = blockIdx.x * TX, y0 = blockIdx.y * TY, z0 = blockIdx.z * TZ;
    const int tid = threadIdx.y * TX + threadIdx.x;
    tile_load_async(u, 0, x0, y0, z0, tid);
    tile_load_async(v, 1, x0, y0, z0, tid);
    tile_load_async(w, 2, x0, y0, z0, tid);
    wait_async();
    __syncthreads();
    tile_patch_bcu(sm, 0, x0, (float)ubp[0], tid);
    __syncthreads();
    const float rdt = frcp(dtp[0]);
    const float* Su = sm;
    const float* Sv = sm + TILE_VOL;
    const float* Sw = sm + 2 * TILE_VOL;
    const int lx = threadIdx.x + XOFF, ly = threadIdx.y + 1;
    for (int zz = 0; zz < TZ; ++zz) {
        const int ci = ((zz + 1) * LY + ly) * LXP + lx;
        const int gi = ((z0 + zz) * N + (y0 + threadIdx.y)) * N + (x0 + threadIdx.x);
        b[gi] = -(ddx(Su, ci) + ddy(Sv, ci) + ddz(Sw, ci)) * rdt;
    }
}

// ---- fused residual + restrict(128->64) + restrict(64->32):
// r = A(edge_pad(p)) - b is never needed at 128^3 by the reference (r_s[0] unused),
// so it stays in registers; block-local LDS quad reductions emit r1 and r2.
__global__ void __launch_bounds__(256)
k_res_restrict(const float* __restrict__ p, const float* __restrict__ b,
               float* __restrict__ r1, float* __restrict__ r2) {
    extern __shared__ float sm[];
    __shared__ float srd[2][TY][TX];
    __shared__ float srd2[2][TY / 2][TX / 2];
    const int x0 = blockIdx.x * TX, y0 = blockIdx.y * TY, z0 = blockIdx.z * TZ;
    const int tx = threadIdx.x, ty = threadIdx.y;
    const int tid = ty * TX + tx;
    tile_load_async(p, 0, x0, y0, z0, tid);
    wait_async();
    __syncthreads();
    const int lx = tx + XOFF, ly = ty + 1;
    float rv[TZ];
    for (int zz = 0; zz < TZ; ++zz) {
        const int ci = ((zz + 1) * LY + ly) * LXP + lx;
        const int gi = ((z0 + zz) * N + (y0 + ty)) * N + (x0 + tx);
        rv[zz] = lap3(sm, ci) - b[gi];
    }
    srd[0][ty][tx] = rv[0] + rv[1];
    srd[1][ty][tx] = rv[2] + rv[3];
    __syncthreads();
    if (((tx | ty) & 1) == 0) {
        for (int k = 0; k < 2; ++k) {
            float s = srd[k][ty][tx] + srd[k][ty][tx + 1] +
                      srd[k][ty + 1][tx] + srd[k][ty + 1][tx + 1];
            float val = 0.125f * s;
            int cz = blockIdx.z * 2 + k;
            int cy = blockIdx.y * 4 + (ty >> 1);
            int cx = blockIdx.x * 16 + (tx >> 1);
            r1[(cz * 64 + cy) * 64 + cx] = val;
            srd2[k][ty >> 1][tx >> 1] = val;
        }
    }
    __syncthreads();
    if (((tx | ty) & 3) == 0) {
        float s = 0.0f;
        int ry = ty >> 1, rx = tx >> 1;
        for (int k = 0; k < 2; ++k)
            for (int dy2 = 0; dy2 < 2; ++dy2)
                for (int dx2 = 0; dx2 < 2; ++dx2)
                    s += srd2[k][ry + dy2][rx + dx2];
        int cz2 = blockIdx.z;
        int cy2 = blockIdx.y * 2 + (ty >> 2);
        int cx2 = blockIdx.x * 8 + (tx >> 2);
        r2[(cz2 * 32 + cy2) * 32 + cx2] = 0.125f * s;
    }
}

// coarse Jacobi-correct + NN-prolong value at fine cell with coarse coords (cx,cy,cz)
__device__ __forceinline__ float mgup_c(const float* __restrict__ win, int nc,
                                        const float* __restrict__ r,
                                        int cx, int cy, int cz) {
    int c = (cz * nc + cy) * nc + cx;
    float wcc = win[c];
    float s = 0.0f;
    if (cx > 0)      s += win[c - 1];
    if (cx < nc - 1) s += win[c + 1];
    if (cy > 0)      s += win[c - nc];
    if (cy < nc - 1) s += win[c + nc];
    if (cz > 0)      s += win[c - nc * nc];
    if (cz < nc - 1) s += win[c + nc * nc];
    return wcc + ((s - 6.0f * wcc) - r[c]) * (1.0f / 6.0f);
}

__device__ __forceinline__ float mgup_val(const float* __restrict__ win, int nc,
                                          const float* __restrict__ r, int idx) {
    const int nf = nc << 1;
    int fx = idx % nf;
    int t  = idx / nf;
    int fy = t % nf;
    int fz = t / nf;
    return mgup_c(win, nc, r, fx >> 1, fy >> 1, fz >> 1);
}

__global__ void __launch_bounds__(256)
k_mg_up(const float* __restrict__ win, const float* __restrict__ r,
        float* __restrict__ out, int nc) {
    const int nf = nc << 1;
    const int tot = nf * nf * nf;
    int idx = blockIdx.x * 256 + threadIdx.x;
    if (idx >= tot) return;
    out[idx] = mgup_val(win, nc, r, idx);
}

// ------ single-block fused MG tail: restrict 32^3 -> 1^3, up 1^3 -> 32^3 -----
__device__ __forceinline__ void restrict_lds(const float* __restrict__ src, int nin,
                                             float* __restrict__ dst,
                                             int tid, int nth) {
    const int nout = nin >> 1;
    const int tot = nout * nout * nout;
    for (int i = tid; i < tot; i += nth) {
        int x = i % nout;
        int t = i / nout;
        int y = t % nout;
        int z = t / nout;
        int xi = 2 * x, yi = 2 * y, zi = 2 * z;
        float s = 0.0f;
        for (int dz2 = 0; dz2 < 2; ++dz2)
            for (int dy2 = 0; dy2 < 2; ++dy2)
                for (int dx2 = 0; dx2 < 2; ++dx2)
                    s += src[((zi + dz2) * nin + (yi + dy2)) * nin + (xi + dx2)];
        dst[i] = 0.125f * s;
    }
}

__global__ void __launch_bounds__(1024)
k_mg_coarse(const float* __restrict__ r2, float* __restrict__ w3,
            float* __restrict__ out_r) {
    __shared__ float rs3[4096];   // 16^3
    __shared__ float rs4[512];    // 8^3
    __shared__ float rs5[64];     // 4^3
    __shared__ float rs6[8];      // 2^3
    __shared__ float rs7[1];      // 1^3
    __shared__ float wA[4096];
    __shared__ float wB[512];
    const int tid = threadIdx.x;
    const int NTH = 1024;
    for (int i = tid; i < 4096; i += NTH) {
        int x = i & 15, t = i >> 4, y = t & 15, z = t >> 4;
        int xi = 2 * x, yi = 2 * y, zi = 2 * z;
        float s = 0.0f;
        for (int dz2 = 0; dz2 < 2; ++dz2)
            for (int dy2 = 0; dy2 < 2; ++dy2)
                for (int dx2 = 0; dx2 < 2; ++dx2)
                    s += r2[((zi + dz2) * 32 + (yi + dy2)) * 32 + (xi + dx2)];
        rs3[i] = 0.125f * s;
    }
    __syncthreads();
    restrict_lds(rs3, 16, rs4, tid, NTH); __syncthreads();
    restrict_lds(rs4,  8, rs5, tid, NTH); __syncthreads();
    restrict_lds(rs5,  4, rs6, tid, NTH); __syncthreads();
    restrict_lds(rs6,  2, rs7, tid, NTH); __syncthreads();
    if (tid == 0) out_r[0] = rs7[0];         // coarsest residual (reference output r)
    if (tid < 8) wB[tid] = -rs7[0] * (1.0f / 6.0f);   // level 7: w=0 => -r/diag
    __syncthreads();
    for (int i = tid; i < 64; i += NTH)    wA[i] = mgup_val(wB, 2, rs6, i);   // -> 4^3
    __syncthreads();
    for (int i = tid; i < 512; i += NTH)   wB[i] = mgup_val(wA, 4, rs5, i);   // -> 8^3
    __syncthreads();
    for (int i = tid; i < 4096; i += NTH)  wA[i] = mgup_val(wB, 8, rs4, i);   // -> 16^3
    __syncthreads();
    for (int i = tid; i < 32768; i += NTH) w3[i] = mgup_val(wA, 16, rs3, i);  // -> 32^3
}

// ------ fused final prolongation + p update:
//   wmg(fine) = mgup(win64, r1) ; p' = p - wmg - A(edge_pad(p))/diag + b/diag
__global__ void __launch_bounds__(256)
k_pupdate(const float* __restrict__ p, const float* __restrict__ win,
          const float* __restrict__ r1, const float* __restrict__ b,
          float* __restrict__ wmg_out, float* __restrict__ pout) {
    extern __shared__ float sm[];
    const int x0 = blockIdx.x * TX, y0 = blockIdx.y * TY, z0 = blockIdx.z * TZ;
    const int tid = threadIdx.y * TX + threadIdx.x;
    tile_load_async(p, 0, x0, y0, z0, tid);
    wait_async();
    __syncthreads();
    const int lx = threadIdx.x + XOFF, ly = threadIdx.y + 1;
    const int gx = x0 + threadIdx.x, gy = y0 + threadIdx.y;
    for (int zz = 0; zz < TZ; ++zz) {
        const int gz = z0 + zz;
        const int ci = ((zz + 1) * LY + ly) * LXP + lx;
        const int gi = (gz * N + gy) * N + gx;
        float wf = mgup_c(win, 64, r1, gx >> 1, gy >> 1, gz >> 1);
        wmg_out[gi] = wf;
        pout[gi] = sm[ci] - wf + (lap3(sm, ci) - b[gi]) * (1.0f / 6.0f);
    }
}

// ------ projection + final solid filter into d_out u,v,w ------
__global__ void __launch_bounds__(256)
k_finalvel(const float* __restrict__ u, const float* __restrict__ v,
           const float* __restrict__ w, const float* __restrict__ p,
           const float* __restrict__ sig, const float* __restrict__ dtp,
           float* __restrict__ uo, float* __restrict__ vo, float* __restrict__ wo) {
    extern __shared__ float sm[];
    const int x0 = blockIdx.x * TX, y0 = blockIdx.y * TY, z0 = blockIdx.z * TZ;
    const int tid = threadIdx.y * TX + threadIdx.x;
    tile_load_async(p, 0, x0, y0, z0, tid);
    wait_async();
    __syncthreads();
    const float dt = dtp[0];
    const int lx = threadIdx.x + XOFF, ly = threadIdx.y + 1;
    for (int zz = 0; zz < TZ; ++zz) {
        const int ci = ((zz + 1) * LY + ly) * LXP + lx;
        const int gi = ((z0 + zz) * N + (y0 + threadIdx.y)) * N + (x0 + threadIdx.x);
        float f = frcp(1.0f + dt * sig[gi]);
        uo[gi] = (u[gi] - dt * ddx(sm, ci)) * f;
        vo[gi] = (v[gi] - dt * ddy(sm, ci)) * f;
        wo[gi] = (w[gi] - dt * ddz(sm, ci)) * f;
    }
}

extern "C" void kernel_launch(void* const* d_in, const int* in_sizes, int n_in,
                              void* d_out, int out_size, void* d_ws, size_t ws_size,
                              hipStream_t stream) {
    (void)in_sizes; (void)n_in; (void)out_size; (void)ws_size;
    const float* in_u = (const float*)d_in[0];
    const float* in_v = (const float*)d_in[1];
    const float* in_w = (const float*)d_in[2];
    const float* in_p = (const float*)d_in[3];
    const float* sig  = (const float*)d_in[4];
    const float* dtp  = (const float*)d_in[5];
    const int*   ubp  = (const int*)d_in[12];
    const int*   rep  = (const int*)d_in[13];
    // iteration=2, nlevel=9 are static hyperparameters of this problem instance.

    float* out_u   = (float*)d_out;
    float* out_v   = out_u + (size_t)NT;
    float* out_w   = out_u + 2 * (size_t)NT;
    float* out_p   = out_u + 3 * (size_t)NT;
    float* out_wmg = out_u + 4 * (size_t)NT;
    float* out_r   = out_u + 5 * (size_t)NT;

    float* ws = (float*)d_ws;
    float* u0 = ws;
    float* v0 = ws + (size_t)NT;
    float* w0 = ws + 2 * (size_t)NT;
    float* b  = ws + 3 * (size_t)NT;
    float* pw = ws + 4 * (size_t)NT;
    float* r1 = ws + 5 * (size_t)NT;   // 64^3 = 262144
    float* r2 = r1 + 262144;           // 32^3 =  32768
    float* w3 = r2 + 32768;            // up-chain 32^3 output of fused MG tail
    float* wb = w3 + 32768;            // up-chain 64^3 buffer (262144)

    // b-field scratch lives in d_out u/v/w slots (overwritten by k_finalvel).
    float* bu = out_u; float* bv = out_v; float* bw = out_w;

    const dim3 blk(TX, TY, 1);
    const dim3 grd(N / TX, N / TY, N / TZ);
    const size_t smem1 = (size_t)TILE_VOL * 4;   //  9600 B
    const size_t smem3 = 3 * smem1;              // 28800 B
    const size_t smem4 = 4 * smem1;              // 38400 B
    const size_t smem5 = 5 * smem1;              // 48000 B

    k_predictor<<<grd, blk, smem5, stream>>>(in_u, in_v, in_w, in_p, sig, dtp, ubp, rep,
                                             bu, bv, bw, u0, v0, w0);
    k_corrector<<<grd, blk, smem4, stream>>>(bu, bv, bw, in_p, sig, dtp, ubp, rep,
                                             u0, v0, w0);
    k_divergence<<<grd, blk, smem3, stream>>>(u0, v0, w0, dtp, ubp, b);

    const float* pin = in_p;
    for (int it = 0; it < 2; ++it) {
        float* pout = (it == 0) ? pw : out_p;
        // residual + restrict to 64^3 and 32^3, all fused
        k_res_restrict<<<grd, blk, smem1, stream>>>(pin, b, r1, r2);
        // MG tail 32^3 -> 1^3 -> 32^3 in one block (LDS resident)
        k_mg_coarse<<<1, 1024, 0, stream>>>(r2, w3, out_r);
        // up-step 32^3 -> 64^3
        k_mg_up<<<(262144 + 255) / 256, 256, 0, stream>>>(w3, r2, wb, 32);
        // fused final prolongation (64^3 -> 128^3 wmg) + p update
        k_pupdate<<<grd, blk, smem1, stream>>>(pin, wb, r1, b, out_wmg, pout);
        pin = pout;
    }

    k_finalvel<<<grd, blk, smem1, stream>>>(u0, v0, w0, pin, sig, dtp, out_u, out_v, out_w);
}